// SimpleCNF_13941463843310
// MI455X (gfx1250) — compile-verified
//
#include <hip/hip_runtime.h>
#include <hip/hip_bf16.h>

// ---------------------------------------------------------------------------
// CNF reverse-time integration, fixed-step RK4.
//   NSTEPS steps of width h = 1/NSTEPS; stage times form a grid of
//   T_GRID = 2*NSTEPS+1 points  s_k = k*h/2,  t_k = 1 - s_k.
// Hypernet evaluated for ALL grid times as two WMMA GEMMs (full fp32,
// V_WMMA_F32_16X16X4_F32). The batch sweep keeps state in registers and
// double-buffers per-step params in LDS via async global->LDS copies
// (ASYNCcnt) so the L2 fetch overlaps the RK4 VALU work.
// ---------------------------------------------------------------------------

#define NSTEPS  40
#define T_GRID  (2 * NSTEPS + 1)          // 81
#define NCOL    96                        // T_GRID padded to 6 N-tiles of 16
#define HIDDEN  64
#define FC3_OUT 448                       // 3*128 + 64
#define PARAM_STRIDE 384                  // 6 arrays of 64 floats per time
#define STEP_FLOATS  (3 * PARAM_STRIDE)   // 1152 floats = 4608 B per step
#define STEP_CHUNKS  (STEP_FLOATS * 4 / 16) // 288 x 16B

typedef __attribute__((ext_vector_type(2))) float v2f;
typedef __attribute__((ext_vector_type(8))) float v8f;
typedef int v4i __attribute__((vector_size(16)));   // matches builtin param

#if defined(__AMDGCN__) && __has_builtin(__builtin_amdgcn_global_load_async_to_lds_b128)
#define HAVE_ASYNC_LDS 1
#else
#define HAVE_ASYNC_LDS 0
#endif

__device__ __forceinline__ void async_wait_all() {
#if HAVE_ASYNC_LDS
#if __has_builtin(__builtin_amdgcn_s_wait_asynccnt)
    __builtin_amdgcn_s_wait_asynccnt(0);
#else
    asm volatile("s_wait_asynccnt 0" ::: "memory");
#endif
#endif
}

// Issue (or synchronously perform) the copy of one step's 4608 B of params
// from global into an LDS buffer. 256 threads participate (288 16B chunks).
__device__ __forceinline__ void stage_params(const float* __restrict__ gsrc,
                                             float* ldst, int tid) {
#if HAVE_ASYNC_LDS
    typedef __attribute__((address_space(1))) v4i GV4;   // target AS1 == __device__
    typedef __attribute__((address_space(3))) v4i LV4;   // target AS3 == __shared__
    GV4* g = (GV4*)(uintptr_t)gsrc;          // global flat addr == AS1 addr
    LV4* l = (LV4*)ldst;                     // addrspacecast generic->LDS
    __builtin_amdgcn_global_load_async_to_lds_b128(g + tid, l + tid, 0, 0);
    if (tid < STEP_CHUNKS - 256)             // remaining 32 chunks
        __builtin_amdgcn_global_load_async_to_lds_b128(g + 256 + tid, l + 256 + tid, 0, 0);
#else
    for (int jj = tid; jj < STEP_FLOATS; jj += 256) ldst[jj] = gsrc[jj];
#endif
}

// ---------------------------------------------------------------------------
// Kernel H1: P1[k][c] = tanh(fc1_w[k]*t_c + fc1_b[k])   (LDS, 64 x 96)
//            P2       = tanh(fc2_w @ P1 + fc2_b)        (WMMA GEMM, 64x96)
// One workgroup, 256 threads = 8 waves; 24 output tiles -> 3 per wave.
// ---------------------------------------------------------------------------
__global__ void cnf_hyper1(const float* __restrict__ fc1_w,
                           const float* __restrict__ fc1_b,
                           const float* __restrict__ fc2_w,
                           const float* __restrict__ fc2_b,
                           float* __restrict__ P2) {
    __shared__ float P1[HIDDEN * NCOL];
    const int tid = threadIdx.x;
    const float hstep = 1.0f / (float)NSTEPS;

    for (int idx = tid; idx < HIDDEN * NCOL; idx += 256) {
        int k = idx / NCOL, c = idx % NCOL;
        int cc = (c < T_GRID) ? c : (T_GRID - 1);        // pad columns
        float t = 1.0f - (float)cc * (0.5f * hstep);
        P1[idx] = tanhf(fc1_w[k] * t + fc1_b[k]);
    }
    __syncthreads();

    const int lane = tid & 31;
    const int wave = tid >> 5;
    const int half = lane >> 4;      // lanes 16-31 handle K+2 / M+8
    const int l16  = lane & 15;

    for (int t3 = 0; t3 < 3; ++t3) {
        int tile = wave * 3 + t3;                        // 0..23
        int mt = tile / 6, nt = tile % 6;
        int arow = mt * 16 + l16;                        // A: lanes = M
        int col  = nt * 16 + l16;                        // B/D: lanes = N
        v8f acc = {};
        for (int kc = 0; kc < 16; ++kc) {
            int k0 = kc * 4 + half * 2;
            v2f a, b;
            a.x = fc2_w[arow * 64 + k0];
            a.y = fc2_w[arow * 64 + k0 + 1];
            b.x = P1[(k0)     * NCOL + col];
            b.y = P1[(k0 + 1) * NCOL + col];
            acc = __builtin_amdgcn_wmma_f32_16x16x4_f32(
                      false, a, false, b, (short)0, acc, false, false);
        }
        #pragma unroll
        for (int v = 0; v < 8; ++v) {
            int r = mt * 16 + v + half * 8;              // D: VGPR v -> row
            P2[r * NCOL + col] = tanhf(acc[v] + fc2_b[r]);
        }
    }
}

// ---------------------------------------------------------------------------
// Kernel H2: P3 = fc3_w(448x64) @ P2(64x96) + fc3_b      (WMMA GEMM)
// One workgroup; 28x6 = 168 tiles -> 21 per wave.
// ---------------------------------------------------------------------------
__global__ void cnf_hyper2(const float* __restrict__ fc3_w,
                           const float* __restrict__ fc3_b,
                           const float* __restrict__ P2,
                           float* __restrict__ P3) {
    __shared__ float sP2[HIDDEN * NCOL];
    const int tid = threadIdx.x;
    for (int idx = tid; idx < HIDDEN * NCOL; idx += 256) sP2[idx] = P2[idx];
    __syncthreads();

    const int lane = tid & 31;
    const int wave = tid >> 5;
    const int half = lane >> 4;
    const int l16  = lane & 15;

    for (int t3 = 0; t3 < 21; ++t3) {
        int tile = wave * 21 + t3;                       // 0..167
        int mt = tile / 6, nt = tile % 6;
        int arow = mt * 16 + l16;
        int col  = nt * 16 + l16;
        v8f acc = {};
        for (int kc = 0; kc < 16; ++kc) {
            int k0 = kc * 4 + half * 2;
            v2f a, b;
            a.x = fc3_w[arow * 64 + k0];
            a.y = fc3_w[arow * 64 + k0 + 1];
            b.x = sP2[(k0)     * NCOL + col];
            b.y = sP2[(k0 + 1) * NCOL + col];
            acc = __builtin_amdgcn_wmma_f32_16x16x4_f32(
                      false, a, false, b, (short)0, acc, false, false);
        }
        #pragma unroll
        for (int v = 0; v < 8; ++v) {
            int r = mt * 16 + v + half * 8;
            P3[r * NCOL + col] = acc[v] + fc3_b[r];
        }
    }
}

// ---------------------------------------------------------------------------
// Kernel H3: derive per-time params from P3 column k:
//   W[i][d]=p[2i+d], U=p[128+..], G=p[256+..], b=p[384+i]
//   ue = U*sigmoid(G), wu[i] = W[i][0]*ue0 + W[i][1]*ue1
// Layout per time: [w0 | w1 | b | u0 | u1 | wu], 64 floats each.
// ---------------------------------------------------------------------------
__global__ void cnf_derive(const float* __restrict__ P3,
                           float* __restrict__ params) {
    int k = blockIdx.x;          // time grid index 0..T_GRID-1
    int j = threadIdx.x;         // hidden unit 0..63
    float w0 = P3[(2 * j)         * NCOL + k];
    float w1 = P3[(2 * j + 1)     * NCOL + k];
    float u0 = P3[(128 + 2 * j)   * NCOL + k];
    float u1 = P3[(128 + 2 * j+1) * NCOL + k];
    float g0 = P3[(256 + 2 * j)   * NCOL + k];
    float g1 = P3[(256 + 2 * j+1) * NCOL + k];
    float bb = P3[(384 + j)       * NCOL + k];
    float ue0 = u0 / (1.0f + expf(-g0));
    float ue1 = u1 / (1.0f + expf(-g1));
    float* pk = params + k * PARAM_STRIDE;
    pk[j]        = w0;
    pk[64 + j]   = w1;
    pk[128 + j]  = bb;
    pk[192 + j]  = ue0;
    pk[256 + j]  = ue1;
    pk[320 + j]  = w0 * ue0 + w1 * ue1;
}

// ---------------------------------------------------------------------------
// Dynamics eval for one point at one stage (params in LDS; all lanes read the
// same LDS word per iteration -> conflict-free broadcast).
// Returns func = (-dz, -dlogp) = (-dz0/64, -dz1/64, +tr/64).
// ---------------------------------------------------------------------------
__device__ __forceinline__ void cnf_eval(const float* __restrict__ sp,
                                         float z0, float z1,
                                         float& fz0, float& fz1, float& flp) {
    float dz0 = 0.f, dz1 = 0.f, tr = 0.f;
    #pragma unroll 8
    for (int j = 0; j < HIDDEN; ++j) {
        float a  = fmaf(z0, sp[j], fmaf(z1, sp[64 + j], sp[128 + j]));
        float hb = tanhf(a);
        dz0 = fmaf(hb, sp[192 + j], dz0);
        dz1 = fmaf(hb, sp[256 + j], dz1);
        tr  = fmaf(1.0f - hb * hb, sp[320 + j], tr);
    }
    const float inv = 1.0f / (float)HIDDEN;
    fz0 = -dz0 * inv;
    fz1 = -dz1 * inv;
    flp =  tr  * inv;
}

// ---------------------------------------------------------------------------
// Kernel I: RK4 sweep. One point per lane; state in registers for all steps.
// Per step, stages use grid points 2i, 2i+1, 2i+2 (contiguous 1152 floats).
// LDS params double-buffered; step i+1's copy is issued asynchronously right
// after the step-i barrier and retired with s_wait_asynccnt + barrier.
// ---------------------------------------------------------------------------
__global__ void cnf_integrate(const float* __restrict__ x,
                              const float* __restrict__ lp_in,
                              const float* __restrict__ params,
                              float* __restrict__ out_z,
                              float* __restrict__ out_lp,
                              int batch) {
    __shared__ float sp[2][STEP_FLOATS];
    const int tid = threadIdx.x;
    const int p   = blockIdx.x * 256 + tid;
    const bool valid = (p < batch);

    float z0 = valid ? x[2 * p]     : 0.f;
    float z1 = valid ? x[2 * p + 1] : 0.f;
    float lp = valid ? lp_in[p]     : 0.f;

    const float h  = 1.0f / (float)NSTEPS;
    const float hh = 0.5f * h;

    stage_params(params, sp[0], tid);                     // prologue: step 0

    for (int i = 0; i < NSTEPS; ++i) {
        async_wait_all();                                 // own wave's copies done
        __syncthreads();                                  // all waves' copies done,
                                                          // all reads of buf[(i+1)&1] done
        if (i + 1 < NSTEPS)
            stage_params(params + (2 * (i + 1)) * PARAM_STRIDE, sp[(i + 1) & 1], tid);
        if (i + 2 < NSTEPS)                               // warm L2 two steps ahead
            __builtin_prefetch(params + (2 * (i + 2)) * PARAM_STRIDE + tid, 0, 1);

        const float* cur = sp[i & 1];
        float a0, a1, al, b0, b1, bl, c0, c1, cl, d0, d1, dl;
        cnf_eval(cur,                    z0,           z1,           a0, a1, al);
        cnf_eval(cur + PARAM_STRIDE,     z0 + hh * a0, z1 + hh * a1, b0, b1, bl);
        cnf_eval(cur + PARAM_STRIDE,     z0 + hh * b0, z1 + hh * b1, c0, c1, cl);
        cnf_eval(cur + 2 * PARAM_STRIDE, z0 + h  * c0, z1 + h  * c1, d0, d1, dl);

        const float w = h * (1.0f / 6.0f);
        z0 += w * (a0 + 2.0f * (b0 + c0) + d0);
        z1 += w * (a1 + 2.0f * (b1 + c1) + d1);
        lp += w * (al + 2.0f * (bl + cl) + dl);
    }

    if (valid) {
        out_z[2 * p]     = z0;
        out_z[2 * p + 1] = z1;
        out_lp[p]        = lp;
    }
}

// ---------------------------------------------------------------------------
extern "C" void kernel_launch(void* const* d_in, const int* in_sizes, int n_in,
                              void* d_out, int out_size, void* d_ws, size_t ws_size,
                              hipStream_t stream) {
    const float* x      = (const float*)d_in[0];   // (BATCH, 2)
    const float* lp_t1  = (const float*)d_in[1];   // (BATCH, 1)
    const float* fc1_w  = (const float*)d_in[2];   // (64, 1)
    const float* fc1_b  = (const float*)d_in[3];   // (64,)
    const float* fc2_w  = (const float*)d_in[4];   // (64, 64)
    const float* fc2_b  = (const float*)d_in[5];   // (64,)
    const float* fc3_w  = (const float*)d_in[6];   // (448, 64)
    const float* fc3_b  = (const float*)d_in[7];   // (448,)

    const int batch = in_sizes[0] / 2;

    float* ws     = (float*)d_ws;
    float* P2     = ws;                                   // 64  * 96
    float* P3     = P2 + HIDDEN * NCOL;                   // 448 * 96
    float* params = P3 + FC3_OUT * NCOL;                  // 81  * 384

    float* out_z  = (float*)d_out;                        // (BATCH, 2)
    float* out_lp = out_z + 2 * batch;                    // (BATCH, 1)

    cnf_hyper1<<<1, 256, 0, stream>>>(fc1_w, fc1_b, fc2_w, fc2_b, P2);
    cnf_hyper2<<<1, 256, 0, stream>>>(fc3_w, fc3_b, P2, P3);
    cnf_derive<<<T_GRID, HIDDEN, 0, stream>>>(P3, params);

    int blocks = (batch + 255) / 256;
    cnf_integrate<<<blocks, 256, 0, stream>>>(x, lp_t1, params, out_z, out_lp, batch);
}